// Session_Aggregator_26104811225298
// MI455X (gfx1250) — compile-verified
//
#include <hip/hip_runtime.h>

// Graph attention segment-softmax aggregation for MI455X (gfx1250, wave32).
// Score phase uses full-precision V_WMMA_F32_16X16X4_F32 (16 pair-dots per
// wave as the diagonal of a 16x16 accumulator, 25 K=4 steps over D=100).

#define NEG_SLOPE 0.2f
#define EMBED_D   100
#define D4        25   // float4 chunks per row

typedef __attribute__((ext_vector_type(2))) float v2f;
typedef __attribute__((ext_vector_type(8))) float v8f;

// Monotone float -> uint key so unsigned atomicMax == float max (deterministic).
__device__ __forceinline__ unsigned orderKey(float f) {
    unsigned u = __float_as_uint(f);
    return (u & 0x80000000u) ? ~u : (u | 0x80000000u);
}
__device__ __forceinline__ float decodeKey(unsigned k) {
    unsigned u = (k & 0x80000000u) ? (k ^ 0x80000000u) : ~k;
    return __uint_as_float(u);
}

__global__ void k_init(unsigned* __restrict__ mkey, int n) {
    int i = blockIdx.x * blockDim.x + threadIdx.x;
    if (i < n) mkey[i] = 0u;   // below any finite score's key
}

// One wave computes scores for 16 pairs via f32 WMMA.
// A[m,k] = h[seg_m][d0+k] * W[d0+k],  B[k,n] = h[nbr_n][d0+k]; diag(C) = scores.
__global__ void __launch_bounds__(256)
k_scores(const float* __restrict__ hidden,
         const float* __restrict__ W,
         const int*   __restrict__ seg,
         const int*   __restrict__ nbr,
         float*       __restrict__ scores,
         unsigned*    __restrict__ mkey,
         int P)
{
    const int lane = threadIdx.x & 31;
    const int wave = (int)((blockIdx.x * blockDim.x + threadIdx.x) >> 5);
    const int pb   = wave * 16;
    if (pb >= P) return;

    const int half = lane >> 4;          // 0: K=0..1, 1: K=2..3  (A & B layout)
    const int mrow = lane & 15;
    int pm = pb + mrow; if (pm >= P) pm = P - 1;   // clamp tail (writes masked)

    const int si = seg[pm];
    const int ni = nbr[pm];

    // float2-granular pointers; row stride 100 floats = 50 v2f, all 8B-aligned.
    const v2f* __restrict__ hs2 = (const v2f*)hidden + (size_t)si * 50 + half;
    const v2f* __restrict__ hn2 = (const v2f*)hidden + (size_t)ni * 50 + half;
    const v2f* __restrict__ w2  = (const v2f*)W + half;

    v8f c = {};
    #pragma unroll
    for (int k = 0; k < 25; ++k) {       // K-dim: d0 = 4k, covers D=100 exactly
        v2f a = hs2[2 * k];
        v2f b = hn2[2 * k];
        a *= w2[2 * k];                  // fold W into A
        c = __builtin_amdgcn_wmma_f32_16x16x4_f32(
                false, a, false, b, (short)0, c, false, false);
    }

    // Diagonal extraction: lanes 0-7 -> pairs 0-7 (vgpr=lane),
    //                      lanes 24-31 -> pairs 8-15 (vgpr=lane-24).
    const int idx = lane & 7;
    float sc = c[0];
    if (idx == 1) sc = c[1];
    if (idx == 2) sc = c[2];
    if (idx == 3) sc = c[3];
    if (idx == 4) sc = c[4];
    if (idx == 5) sc = c[5];
    if (idx == 6) sc = c[6];
    if (idx == 7) sc = c[7];

    const bool holdsDiag = (lane < 8) || (lane >= 24);
    const int  m = (lane < 8) ? lane : (lane - 16);
    const int  p = pb + m;
    if (holdsDiag && p < P) {
        sc = (sc >= 0.0f) ? sc : NEG_SLOPE * sc;   // leaky relu
        scores[p] = sc;
        atomicMax(&mkey[si], orderKey(sc));        // deterministic integer max
    }
}

// e_p = exp(s_p - m[seg_p]); record segment starts (seg sorted, no empty segs).
__global__ void k_exp(float* __restrict__ sc,
                      const int* __restrict__ seg,
                      const unsigned* __restrict__ mkey,
                      int* __restrict__ rstart,
                      int P)
{
    int p = blockIdx.x * blockDim.x + threadIdx.x;
    if (p >= P) return;
    int v = seg[p];
    float mx = decodeKey(mkey[v]);
    sc[p] = __expf(sc[p] - mx);
    if (p == 0 || seg[p - 1] != v) rstart[v] = p;
}

// One wave per node: out[v] = (sum_p e_p * h[nbr_p]) / (sum_p e_p).
// Serial pair loop -> deterministic; lanes 0-24 each own one float4 of the row.
__global__ void __launch_bounds__(256)
k_aggregate(const float* __restrict__ hidden,
            const int*   __restrict__ nbr,
            const float* __restrict__ e,
            const int*   __restrict__ rstart,
            float*       __restrict__ out,
            int N, int P)
{
    const int lane = threadIdx.x & 31;
    const int v    = (int)((blockIdx.x * blockDim.x + threadIdx.x) >> 5);
    if (v >= N) return;

    const int s    = rstart[v];
    const int send = (v + 1 < N) ? rstart[v + 1] : P;

    float4 acc = make_float4(0.f, 0.f, 0.f, 0.f);
    float  z   = 0.f;
    for (int p = s; p < send; ++p) {
        const float w = e[p];              // uniform across wave
        z += w;
        if (lane < D4) {
            const float4 h =
                *((const float4*)(hidden + (size_t)nbr[p] * EMBED_D) + lane);
            acc.x += w * h.x; acc.y += w * h.y;
            acc.z += w * h.z; acc.w += w * h.w;
        }
    }
    const float rz = 1.0f / z;
    if (lane < D4) {
        acc.x *= rz; acc.y *= rz; acc.z *= rz; acc.w *= rz;
        *((float4*)(out + (size_t)v * EMBED_D) + lane) = acc;
    }
}

extern "C" void kernel_launch(void* const* d_in, const int* in_sizes, int n_in,
                              void* d_out, int out_size, void* d_ws, size_t ws_size,
                              hipStream_t stream)
{
    const float* hidden = (const float*)d_in[0];   // [N, 100] f32
    const float* W      = (const float*)d_in[1];   // [100]    f32
    const int*   seg    = (const int*)  d_in[2];   // [P] sorted
    const int*   nbr    = (const int*)  d_in[3];   // [P]

    const int P = in_sizes[2];
    const int N = in_sizes[0] / EMBED_D;
    float* out  = (float*)d_out;

    // Workspace layout (all 4-byte aligned): scores[P], mkey[N], rstart[N]
    char* ws = (char*)d_ws;
    float*    scores = (float*)ws;
    unsigned* mkey   = (unsigned*)(ws + (size_t)P * 4);
    int*      rstart = (int*)(ws + (size_t)P * 4 + (size_t)N * 4);

    k_init<<<(N + 255) / 256, 256, 0, stream>>>(mkey, N);

    const int waves   = (P + 15) / 16;             // 16 pairs per wave
    const int blocksS = (waves * 32 + 255) / 256;
    k_scores<<<blocksS, 256, 0, stream>>>(hidden, W, seg, nbr, scores, mkey, P);

    k_exp<<<(P + 255) / 256, 256, 0, stream>>>(scores, seg, mkey, rstart, P);

    const int blocksA = (N + 7) / 8;               // 8 waves (nodes) per block
    k_aggregate<<<blocksA, 256, 0, stream>>>(hidden, nbr, scores, rstart, out, N, P);
}